// KBpA_56813827391590
// MI455X (gfx1250) — compile-verified
//
#include <hip/hip_runtime.h>

#define N_PIXELS  262144
#define N_CENTERS 1024
#define N_G       256

typedef float v2f __attribute__((ext_vector_type(2)));
typedef float v8f __attribute__((ext_vector_type(8)));

// Native exp2 on the trans unit (v_exp_f32 computes 2^x).
#if __has_builtin(__builtin_amdgcn_exp2f)
#define EXP2F(x) __builtin_amdgcn_exp2f(x)
#else
#define EXP2F(x) __expf((x) * 0.6931471805599453f)
#endif

// ---------------------------------------------------------------------------
// Kernel 1: deformed = all_pixels - K @ betas
// One wave (32 lanes) per pixel row; K row (1 KB) read as coalesced float4.
// Memory-bound: streams K (256 MB) exactly once -> ~11 us roofline at 23.3 TB/s.
// ---------------------------------------------------------------------------
__global__ __launch_bounds__(256) void deform_kernel(
    const float* __restrict__ K, const float* __restrict__ betas,
    const float* __restrict__ pixels, float* __restrict__ deformed)
{
    __shared__ float sb[2 * N_G];          // betas, 512 floats
    int tid = threadIdx.x;
    sb[tid]       = betas[tid];
    sb[tid + 256] = betas[tid + 256];
    __syncthreads();

    int lane = tid & 31;
    int wave = tid >> 5;
    int p = blockIdx.x * 8 + wave;

    const float4* K4 = (const float4*)(K + (size_t)p * N_G);
    float4 k0 = K4[lane];        // g = lane*4 .. lane*4+3
    float4 k1 = K4[lane + 32];   // g = 128 + lane*4 ..

    float kk0[4] = {k0.x, k0.y, k0.z, k0.w};
    float kk1[4] = {k1.x, k1.y, k1.z, k1.w};

    float ax = 0.0f, ay = 0.0f;
#pragma unroll
    for (int j = 0; j < 4; ++j) {
        int g  = lane * 4 + j;
        int g2 = 128 + lane * 4 + j;
        ax = fmaf(kk0[j], sb[2 * g],      ax);
        ay = fmaf(kk0[j], sb[2 * g + 1],  ay);
        ax = fmaf(kk1[j], sb[2 * g2],     ax);
        ay = fmaf(kk1[j], sb[2 * g2 + 1], ay);
    }
#pragma unroll
    for (int off = 16; off > 0; off >>= 1) {
        ax += __shfl_xor(ax, off, 32);
        ay += __shfl_xor(ay, off, 32);
    }
    if (lane == 0) {
        deformed[2 * p]     = pixels[2 * p]     - ax;
        deformed[2 * p + 1] = pixels[2 * p + 1] - ay;
    }
}

// ---------------------------------------------------------------------------
// Kernel 2: out[p] = sum_c alphas[c] * exp(-(||d_p||^2 + ||c||^2 - 2 d_p.c)
//                                           / (2*sdp2))
// Fused: big_matrix/kBp never hit memory. Each wave owns 16 pixels; the
// 16x16 dot-product tile uses V_WMMA_F32_16X16X4_F32 (K=2 padded to 4).
// exp(x) is computed as exp2 with the scale s2 = -log2(e)/(2*sdp2) folded
// into the pixel/center norms up front, so the per-element work is just
// add + fma + v_exp_f32 + pk_fma. Centers packed as float4 {cx,cy,cn*s2,al}
// in LDS -> one ds_load_b128 per tile per lane.
// ---------------------------------------------------------------------------
__global__ __launch_bounds__(256) void kbp_kernel(
    const float* __restrict__ deformed, const float* __restrict__ centers,
    const float* __restrict__ alphas, const float* __restrict__ sdp2p,
    float* __restrict__ out)
{
    __shared__ float4 s_c[N_CENTERS];   // {cx, cy, |c|^2 * s2, alpha}

    int tid = threadIdx.x;
    // exp(-x/(2*sdp2)) == exp2(x * s2), s2 = -log2(e)/(2*sdp2)
    float s2 = -1.442695040888963f / (2.0f * sdp2p[0]);

    for (int c = tid; c < N_CENTERS; c += 256) {
        float cx = centers[2 * c];
        float cy = centers[2 * c + 1];
        s_c[c] = make_float4(cx, cy, (cx * cx + cy * cy) * s2, alphas[c]);
    }
    __syncthreads();

    int lane = tid & 31;
    int wave = tid >> 5;
    int hi   = lane >> 4;     // 0: lanes 0-15 (K=0,1), 1: lanes 16-31 (K=2,3 pad)
    int l15  = lane & 15;

    int p0 = blockIdx.x * 128 + wave * 16;   // this wave's 16 pixels

    // Lane L mirrors pixel p0 + (L&15); hi lanes are zero-padded in A.
    float dx = deformed[2 * (p0 + l15)];
    float dy = deformed[2 * (p0 + l15) + 1];
    float pnB_local = (dx * dx + dy * dy) * s2;   // pre-scaled pixel norm

    // A matrix (16x4 f32): v[0]=K0, v[1]=K1 on lanes 0-15; K2,K3 = 0.
    v2f a;
    a[0] = hi ? 0.0f : dx;
    a[1] = hi ? 0.0f : dy;

    // Broadcast pre-scaled pixel norms: for D-vgpr v, this lane's row is
    // M = v + 8*hi.
    float pnB[8];
#pragma unroll
    for (int v = 0; v < 8; ++v)
        pnB[v] = __shfl(pnB_local, v + 8 * hi, 32);

    float m2 = -2.0f * s2;   // coefficient of the dot product in the exp2 arg

    float acc[8] = {0, 0, 0, 0, 0, 0, 0, 0};
    v8f cz = {};   // zero C accumulator

    for (int n0 = 0; n0 < N_CENTERS; n0 += 16) {
        float4 c4 = s_c[n0 + l15];           // one ds_load_b128
        // B matrix (4x16 f32): rows K0,K1 on lanes 0-15; K2,K3 = 0.
        v2f b;
        b[0] = hi ? 0.0f : c4.x;
        b[1] = hi ? 0.0f : c4.y;

        // D[m][n] = deformed[p0+m] . centers[n0+n]
        v8f d = __builtin_amdgcn_wmma_f32_16x16x4_f32(
            false, a, false, b, (short)0, cz, false, false);

#pragma unroll
        for (int v = 0; v < 8; ++v) {
            float arg = fmaf(m2, d[v], pnB[v] + c4.z);  // full exp2 argument
            acc[v] = fmaf(c4.w, EXP2F(arg), acc[v]);
        }
    }

    // Reduce each D row across its 16 lanes (xor stays within each half).
#pragma unroll
    for (int v = 0; v < 8; ++v) {
        float s = acc[v];
        s += __shfl_xor(s, 1, 32);
        s += __shfl_xor(s, 2, 32);
        s += __shfl_xor(s, 4, 32);
        s += __shfl_xor(s, 8, 32);
        acc[v] = s;
    }
    if (l15 == 0) {             // lane 0 -> pixels p0..p0+7, lane 16 -> p0+8..15
#pragma unroll
        for (int v = 0; v < 8; ++v)
            out[p0 + v + 8 * hi] = acc[v];
    }
}

// ---------------------------------------------------------------------------
extern "C" void kernel_launch(void* const* d_in, const int* in_sizes, int n_in,
                              void* d_out, int out_size, void* d_ws, size_t ws_size,
                              hipStream_t stream) {
    (void)in_sizes; (void)n_in; (void)out_size; (void)ws_size;
    const float* alphas  = (const float*)d_in[0];   // (1024,1)
    const float* betas   = (const float*)d_in[1];   // (256,2)
    const float* K       = (const float*)d_in[2];   // (262144,256)
    const float* pixels  = (const float*)d_in[3];   // (262144,2)
    const float* centers = (const float*)d_in[4];   // (1024,2)
    const float* sdp2    = (const float*)d_in[5];   // scalar

    float* deformed = (float*)d_ws;                 // 262144*2 floats = 2 MB
    float* out      = (float*)d_out;                // (262144,1)

    deform_kernel<<<N_PIXELS / 8,   256, 0, stream>>>(K, betas, pixels, deformed);
    kbp_kernel   <<<N_PIXELS / 128, 256, 0, stream>>>(deformed, centers, alphas,
                                                      sdp2, out);
}